// Project_28132035789005
// MI455X (gfx1250) — compile-verified
//
#include <hip/hip_runtime.h>

// ---------------------------------------------------------------------------
// Pipeline: project+zbuffer (WMMA f32 16x16x4) -> resolve scatter -> mask ->
//           10x 5x5 median inpaint (LDS-tiled) -> 3x3 any>0 count -> output
// ---------------------------------------------------------------------------

typedef float v2f __attribute__((ext_vector_type(2)));
typedef float v8f __attribute__((ext_vector_type(8)));

constexpr int IMG_H = 240;
constexpr int IMG_W = 320;
constexpr int HW    = IMG_H * IMG_W;
constexpr int TX = 32, TY = 8;           // median tile (blockDim = 256)

// ---------------- init: depth buffer = INT_MAX, image = -0.001 -------------
__global__ void init_kernel(int* __restrict__ depth_buf,
                            float* __restrict__ img, int M3, int M) {
  int i = blockIdx.x * blockDim.x + threadIdx.x;
  if (i < M)  depth_buf[i] = 0x7FFFFFFF;
  if (i < M3) img[i] = -0.001f;
}

// ---------------- projection + z-buffer via V_WMMA_F32_16X16X4_F32 ---------
// A (16x4): rows 0..2 = K @ T[0:3,:], row 3 = T[2,:], rows 4..15 = 0.
// B (4x16): 16 points (x,y,z,w). C rows 0..3 -> (px,py,pz,depth) per point.
__global__ void proj_zbuf_kernel(const float* __restrict__ cloud,
                                 const float* __restrict__ Km,
                                 const float* __restrict__ Tm,
                                 int N,
                                 int* __restrict__ depth_buf,
                                 int* __restrict__ rec_lin,
                                 int* __restrict__ rec_d) {
  const int lane   = threadIdx.x & 31;
  const int waveId = (blockIdx.x * blockDim.x + threadIdx.x) >> 5;
  const int base   = waveId * 16;                 // 16 points per wave
  const int b      = base / N;
  const int n0     = base - b * N;

  const float* Kb = Km + b * 9;
  const float* Tb = Tm + b * 16;

  // A operand: lane holds A[r][c0], A[r][c1] ; r = lane&15, lower half-wave
  // carries columns {0,1}, upper half-wave columns {2,3}.
  const int r  = lane & 15;
  const int c0 = (lane < 16) ? 0 : 2;
  const int c1 = (lane < 16) ? 1 : 3;
  float a0 = 0.0f, a1 = 0.0f;
  if (r < 3) {
    a0 = Kb[r*3+0]*Tb[0*4+c0] + Kb[r*3+1]*Tb[1*4+c0] + Kb[r*3+2]*Tb[2*4+c0];
    a1 = Kb[r*3+0]*Tb[0*4+c1] + Kb[r*3+1]*Tb[1*4+c1] + Kb[r*3+2]*Tb[2*4+c1];
  } else if (r == 3) {
    a0 = Tb[2*4+c0];
    a1 = Tb[2*4+c1];
  }
  v2f am; am.x = a0; am.y = a1;

  // B operand: VGPR0 = rows K=0|K=2, VGPR1 = rows K=1|K=3 across half-waves.
  const int prow0 = (lane < 16) ? 0 : 2;
  const int prow1 = (lane < 16) ? 1 : 3;
  const int pn    = n0 + (lane & 15);
  const float* cb = cloud + (size_t)b * 4 * (size_t)N;
  v2f bm;
  bm.x = cb[(size_t)prow0 * N + pn];
  bm.y = cb[(size_t)prow1 * N + pn];

  v8f acc = {};
  acc = __builtin_amdgcn_wmma_f32_16x16x4_f32(false, am, false, bm,
                                              (short)0, acc, false, false);

  // C layout: lanes 0..15, VGPR m = row m (point = lane).
  if (lane < 16) {
    float px = acc[0], py = acc[1], pz = acc[2], dep = acc[3];
    float uf = px / pz;
    float vf = py / pz;
    int u = (int)uf;
    int v = (int)vf;
    u = u < 0 ? 0 : (u > IMG_W - 1 ? IMG_W - 1 : u);
    v = v < 0 ? 0 : (v > IMG_H - 1 ? IMG_H - 1 : v);
    int lin   = b * HW + v * IMG_W + u;
    int d_int = (int)dep;
    atomicMin(&depth_buf[lin], d_int);
    int pt = base + (lane & 15);
    rec_lin[pt] = lin;
    rec_d[pt]   = d_int;
  }
}

// ---------------- visibility resolve: scatter rgb of z-test winners --------
__global__ void resolve_kernel(const float* __restrict__ rgb,
                               const int* __restrict__ rec_lin,
                               const int* __restrict__ rec_d,
                               const int* __restrict__ depth_buf,
                               float* __restrict__ img, int N, int BN) {
  int i = blockIdx.x * blockDim.x + threadIdx.x;
  if (i >= BN) return;
  int lin = rec_lin[i];
  if (rec_d[i] != depth_buf[lin]) return;
  int b = i / N, n = i - b * N;
  int pix = lin - b * HW;
  float* dst = img + (size_t)(b * 3) * HW + pix;
  const float* src = rgb + (size_t)(b * 3) * N + n;
  dst[0]        = src[0];
  dst[HW]       = src[N];
  dst[2 * HW]   = src[2 * N];
}

// ---------------- inpaint mask from ORIGINAL channel 0 ---------------------
__global__ void mask_kernel(const float* __restrict__ img,
                            unsigned char* __restrict__ mask, int M) {
  int i = blockIdx.x * blockDim.x + threadIdx.x;
  if (i >= M) return;
  int b = i / HW, pix = i - b * HW;
  mask[i] = (img[(size_t)(b * 3) * HW + pix] > 0.0f) ? 1 : 0;
}

// ---------------- 5x5 median inpaint (reflect pad), LDS-tiled --------------
__device__ __forceinline__ int reflect(int i, int n) {
  return i < 0 ? -i : (i >= n ? 2 * n - 2 - i : i);
}

__global__ void median_kernel(const float* __restrict__ in,
                              const unsigned char* __restrict__ mask,
                              float* __restrict__ out) {
  __shared__ float tile[TY + 4][TX + 4];
  const int plane = blockIdx.z;           // b*3 + c
  const int b     = plane / 3;
  const int bx0   = blockIdx.x * TX;
  const int by0   = blockIdx.y * TY;
  const float* pin = in + (size_t)plane * HW;
  const int tid = threadIdx.x;            // 0..255

  for (int idx = tid; idx < (TY + 4) * (TX + 4); idx += 256) {
    int ly = idx / (TX + 4);
    int lx = idx - ly * (TX + 4);
    int gy = reflect(by0 + ly - 2, IMG_H);
    int gx = reflect(bx0 + lx - 2, IMG_W);
    tile[ly][lx] = pin[gy * IMG_W + gx];
  }
  __syncthreads();

  const int tx = tid & 31, ty = tid >> 5;
  const int gx = bx0 + tx, gy = by0 + ty;
  float result;
  if (mask[b * HW + gy * IMG_W + gx]) {
    result = tile[ty + 2][tx + 2];
  } else {
    float w[25];
#pragma unroll
    for (int dy = 0; dy < 5; ++dy)
#pragma unroll
      for (int dx = 0; dx < 5; ++dx)
        w[dy * 5 + dx] = tile[ty + dy][tx + dx];
    // partial selection network: after pass i, w[i] = i-th smallest.
#pragma unroll
    for (int i = 0; i < 13; ++i)
#pragma unroll
      for (int j = i + 1; j < 25; ++j) {
        float lo = fminf(w[i], w[j]);
        float hi = fmaxf(w[i], w[j]);
        w[i] = lo; w[j] = hi;
      }
    result = w[12];                       // exact median of 25
  }
  out[(size_t)plane * HW + gy * IMG_W + gx] = result;
}

// ---------------- any(channel > 0) per pixel -------------------------------
__global__ void anypos_kernel(const float* __restrict__ img,
                              unsigned char* __restrict__ ap, int M) {
  int i = blockIdx.x * blockDim.x + threadIdx.x;
  if (i >= M) return;
  int b = i / HW, pix = i - b * HW;
  const float* p = img + (size_t)(b * 3) * HW + pix;
  ap[i] = (unsigned char)((p[0] > 0.0f) | (p[HW] > 0.0f) | (p[2 * HW] > 0.0f));
}

// ---------------- 3x3 neighbor count (zero pad), threshold, write out ------
__global__ void hole_kernel(const float* __restrict__ img,
                            const unsigned char* __restrict__ ap,
                            float* __restrict__ out, int M) {
  int i = blockIdx.x * blockDim.x + threadIdx.x;
  if (i >= M) return;
  int b = i / HW, pix = i - b * HW;
  int h = pix / IMG_W, w = pix - h * IMG_W;
  int cnt = 0;
#pragma unroll
  for (int dy = -1; dy <= 1; ++dy)
#pragma unroll
    for (int dx = -1; dx <= 1; ++dx) {
      int y = h + dy, x = w + dx;
      if (y >= 0 && y < IMG_H && x >= 0 && x < IMG_W)
        cnt += ap[b * HW + y * IMG_W + x];
    }
  float hole = (cnt >= 6) ? 1.0f : 0.0f;
  const float* p = img + (size_t)(b * 3) * HW + pix;
  float* o = out + (size_t)(b * 3) * HW + pix;
  o[0]      = p[0]      * hole;
  o[HW]     = p[HW]     * hole;
  o[2 * HW] = p[2 * HW] * hole;
}

// ---------------------------------------------------------------------------
extern "C" void kernel_launch(void* const* d_in, const int* in_sizes, int n_in,
                              void* d_out, int out_size, void* d_ws, size_t ws_size,
                              hipStream_t stream) {
  const float* cloud = (const float*)d_in[0];   // (B,4,N)
  const float* rgb   = (const float*)d_in[1];   // (B,3,N)
  const float* Km    = (const float*)d_in[2];   // (B,3,3)
  const float* Tm    = (const float*)d_in[3];   // (B,4,4)
  float* out = (float*)d_out;                   // (B,3,H,W)

  const int B  = in_sizes[2] / 9;
  const int N  = in_sizes[0] / (4 * B);
  const int BN = B * N;
  const int M  = B * HW;

  // Workspace layout (256B-aligned blocks).
  char* ws = (char*)d_ws;
  auto al = [](size_t x) { return (x + 255) & ~(size_t)255; };
  size_t off = 0;
  int* depth_buf = (int*)(ws + off);            off += al((size_t)M * 4);
  int* rec_lin   = (int*)(ws + off);            off += al((size_t)BN * 4);
  int* rec_d     = (int*)(ws + off);            off += al((size_t)BN * 4);
  float* imgA    = (float*)(ws + off);          off += al((size_t)3 * M * 4);
  float* imgB    = (float*)(ws + off);          off += al((size_t)3 * M * 4);
  unsigned char* mask = (unsigned char*)(ws + off); off += al((size_t)M);
  unsigned char* ap   = (unsigned char*)(ws + off);

  init_kernel<<<(3 * M + 255) / 256, 256, 0, stream>>>(depth_buf, imgA, 3 * M, M);

  // 16 points per wave, 8 waves per 256-thread block; BN = 524288 divides exactly.
  const int waves  = BN / 16;
  const int blocks = (waves * 32) / 256;
  proj_zbuf_kernel<<<blocks, 256, 0, stream>>>(cloud, Km, Tm, N,
                                               depth_buf, rec_lin, rec_d);
  resolve_kernel<<<(BN + 255) / 256, 256, 0, stream>>>(rgb, rec_lin, rec_d,
                                                       depth_buf, imgA, N, BN);
  mask_kernel<<<(M + 255) / 256, 256, 0, stream>>>(imgA, mask, M);

  dim3 mg(IMG_W / TX, IMG_H / TY, (unsigned)(B * 3));
  float* pin = imgA;
  float* pout = imgB;
  for (int it = 0; it < 10; ++it) {
    median_kernel<<<mg, 256, 0, stream>>>(pin, mask, pout);
    float* t = pin; pin = pout; pout = t;
  }
  // after 10 swaps, result is back in imgA (== pin)

  anypos_kernel<<<(M + 255) / 256, 256, 0, stream>>>(pin, ap, M);
  hole_kernel<<<(M + 255) / 256, 256, 0, stream>>>(pin, ap, out, M);
}